// SoftmaxRefMatcher_86079734546636
// MI455X (gfx1250) — compile-verified
//
#include <hip/hip_runtime.h>
#include <hip/hip_bf16.h>
#include <math.h>

typedef __attribute__((ext_vector_type(2))) float v2f;
typedef __attribute__((ext_vector_type(8))) float v8f;

#define WIDTH 320
#define HW (WIDTH * WIDTH)        // 102400
#define CCH 256
#define NKP 1024
#define NPAIR 2
#define NCHUNK 50
#define CHUNK_PX 2048
#define TILES_PER_CHUNK (CHUNK_PX / 16)   // 128
#define QG_PER_PAIR 8                      // 1024 / (8 waves * 16 q)
#define EPSN 1e-12f

// workspace layout in floats
#define WS_QN 0
#define WS_INVK (WS_QN + NPAIR * NKP * CCH)   // 524288
#define WS_PART (WS_INVK + NPAIR * HW)        // 729088 (+ 2*1024*50*4)

__device__ __forceinline__ float rmax16(float x) {
    x = fmaxf(x, __shfl_xor(x, 1, 16));
    x = fmaxf(x, __shfl_xor(x, 2, 16));
    x = fmaxf(x, __shfl_xor(x, 4, 16));
    x = fmaxf(x, __shfl_xor(x, 8, 16));
    return x;
}
__device__ __forceinline__ float rsum16(float x) {
    x += __shfl_xor(x, 1, 16);
    x += __shfl_xor(x, 2, 16);
    x += __shfl_xor(x, 4, 16);
    x += __shfl_xor(x, 8, 16);
    return x;
}

// issue one per-lane async global->LDS 16-byte copy (CDNA5 GLOBAL_LOAD_ASYNC_TO_LDS_B128)
__device__ __forceinline__ void async_ld_b128(unsigned lds_off, unsigned g_off, unsigned long long gbase) {
    asm volatile("global_load_async_to_lds_b128 %0, %1, %2"
                 :: "v"(lds_off), "v"(g_off), "s"(gbase) : "memory");
}
__device__ __forceinline__ void wait_async0() {
    asm volatile("s_wait_asynccnt 0x0" ::: "memory");
}

// Kernel 1: normalize target keypoint descriptors -> qn[p][n][c]
__global__ void qnorm_kernel(const float* __restrict__ kdesc, float* __restrict__ qn) {
    int gid = blockIdx.x * 256 + threadIdx.x;       // 0..2047
    int p = gid / NKP, n = gid % NKP;
    const float* base = kdesc + ((size_t)(2 * p + 1) * CCH) * NKP + n;
    float ss = 0.f;
    for (int c = 0; c < CCH; ++c) { float x = base[(size_t)c * NKP]; ss += x * x; }
    float inv = 1.0f / fmaxf(sqrtf(ss), EPSN);
    float* o = qn + (size_t)gid * CCH;
    for (int c = 0; c < CCH; ++c) o[c] = base[(size_t)c * NKP] * inv;
}

// Kernel 2: per-pixel inverse L2 norm of dense source descriptors
__global__ void knorm_kernel(const float* __restrict__ dense, float* __restrict__ invk) {
    int gid = blockIdx.x * 256 + threadIdx.x;       // 0..204799
    int p = gid / HW, px = gid % HW;
    const float* base = dense + ((size_t)(2 * p) * CCH) * HW + px;
    float ss = 0.f;
    for (int c = 0; c < CCH; ++c) { float x = base[(size_t)c * HW]; ss += x * x; }
    invk[gid] = 1.0f / fmaxf(sqrtf(ss), EPSN);
}

// Kernel 3: flash softmax-argmax. grid = P * QG_PER_PAIR * NCHUNK blocks, 256 threads.
__global__ __launch_bounds__(256) void matcher_main(
    const float* __restrict__ dense, const float* __restrict__ qn,
    const float* __restrict__ invk, float* __restrict__ part)
{
    __shared__ float lds_k[2][CCH * 16];            // 2 x 16 KB double-buffered K-tile, [c][px]
    const int b = blockIdx.x;
    const int p = b / (QG_PER_PAIR * NCHUNK);
    const int qg = (b % (QG_PER_PAIR * NCHUNK)) / NCHUNK;
    const int chunk = b % NCHUNK;
    const int tid = threadIdx.x;
    const int wave = tid >> 5, lane = tid & 31;
    const int half = lane >> 4, nn = lane & 15;
    const int qbase = qg * 128 + wave * 16;
    const long pxchunk = (long)chunk * CHUNK_PX;

    // A fragments: 16 queries x 256 channels kept in VGPRs (64 x float2 per lane).
    // fp32 A layout: lanes 0-15 row M, VGPR0/1 = K0/K1; lanes 16-31 = K2/K3.
    v2f q[64];
    const float* qrow = qn + ((size_t)(p * NKP + qbase + nn)) * CCH + 2 * half;
#pragma unroll
    for (int s = 0; s < 64; ++s) q[s] = *(const v2f*)(qrow + 4 * s);

    float rowmax[8], rowsum[8], rowu[8], rowv[8];
#pragma unroll
    for (int r = 0; r < 8; ++r) { rowmax[r] = -3.0e38f; rowsum[r] = 0.f; rowu[r] = 0.f; rowv[r] = 0.f; }

    const float* kbase = dense + ((size_t)(2 * p) * CCH) * HW;
    const float* ikp = invk + (size_t)p * HW;

    // async staging setup: thread tid copies channel c = tid (16 floats = 4 x b128)
    const int c = tid;
    const unsigned long long gbase = (unsigned long long)(uintptr_t)kbase;
    const unsigned g_ch_off = (unsigned)((size_t)c * HW * sizeof(float));   // < 105 MB, fits i32
    unsigned ldsoff[2];
    ldsoff[0] = (unsigned)(uintptr_t)&lds_k[0][c * 16];  // low 32 bits of generic addr = LDS offset
    ldsoff[1] = (unsigned)(uintptr_t)&lds_k[1][c * 16];

    // prologue: fill buffer 0 with tile 0
    {
        unsigned go = g_ch_off + (unsigned)(pxchunk * sizeof(float));
#pragma unroll
        for (int j = 0; j < 4; ++j) async_ld_b128(ldsoff[0] + j * 16, go + j * 16, gbase);
    }

    for (int t = 0; t < TILES_PER_CHUNK; ++t) {
        const int bufi = t & 1;
        const long px0 = pxchunk + (long)t * 16;

        wait_async0();          // this wave's fill of buf[bufi] complete
        __syncthreads();        // all waves' fills visible; all done reading buf[bufi^1]

        if (t + 1 < TILES_PER_CHUNK) {   // fire-and-forget fill of next tile
            unsigned go = g_ch_off + (unsigned)((px0 + 16) * sizeof(float));
#pragma unroll
            for (int j = 0; j < 4; ++j) async_ld_b128(ldsoff[bufi ^ 1] + j * 16, go + j * 16, gbase);
        }

        // 16x16 score tile: 64 steps of v_wmma_f32_16x16x4_f32 over C=256
        const float* lk = &lds_k[bufi][0];
        v8f acc = {0.f, 0.f, 0.f, 0.f, 0.f, 0.f, 0.f, 0.f};
#pragma unroll
        for (int s = 0; s < 64; ++s) {
            const int c0 = 4 * s + 2 * half;        // B layout: VGPR v, lane L: K = v + 2*(L/16)
            v2f bf;
            bf.x = lk[c0 * 16 + nn];
            bf.y = lk[(c0 + 1) * 16 + nn];
            acc = __builtin_amdgcn_wmma_f32_16x16x4_f32(
                false, q[s], false, bf, (short)0, acc, false, false);
        }

        const int px = (int)px0 + nn;
        const float ik = ikp[px];
        const float uu = (float)(px % WIDTH);
        const float vv = (float)(px / WIDTH);

        // online softmax + soft-argmax accumulation; C/D layout: VGPR r, lane L:
        // row M = r + 8*(L/16), col N = L%16
#pragma unroll
        for (int r = 0; r < 8; ++r) {
            float s  = acc[r] * ik * 100.0f;        // 1/temperature
            float tm = rmax16(s);
            float nm = fmaxf(rowmax[r], tm);
            float corr = __expf(rowmax[r] - nm);
            float pe = __expf(s - nm);
            rowsum[r] = rowsum[r] * corr + rsum16(pe);
            rowu[r]   = rowu[r]   * corr + rsum16(pe * uu);
            rowv[r]   = rowv[r]   * corr + rsum16(pe * vv);
            rowmax[r] = nm;
        }
    }

    // write chunk partials (max, sum, su, sv) once per row
#pragma unroll
    for (int r = 0; r < 8; ++r) {
        if (nn == r) {
            int m = r + 8 * half;
            int qi = qbase + m;
            float4 o = make_float4(rowmax[r], rowsum[r], rowu[r], rowv[r]);
            *(float4*)&part[(((size_t)(p * NKP + qi)) * NCHUNK + chunk) * 4] = o;
        }
    }
}

// Kernel 4: merge chunk partials, emit outputs
__global__ void matcher_reduce(const float* __restrict__ part,
                               const float* __restrict__ kp_scores,
                               float* __restrict__ out) {
    int gid = blockIdx.x * 256 + threadIdx.x;       // 0..2047  (p*1024 + q)
    int p = gid / NKP, qv = gid % NKP;
    const float* pp = part + (size_t)gid * NCHUNK * 4;
    float g = -3.0e38f;
    for (int ch = 0; ch < NCHUNK; ++ch) g = fmaxf(g, pp[ch * 4]);
    float l = 0.f, U = 0.f, V = 0.f;
    for (int ch = 0; ch < NCHUNK; ++ch) {
        float w = __expf(pp[ch * 4] - g);
        l += pp[ch * 4 + 1] * w;
        U += pp[ch * 4 + 2] * w;
        V += pp[ch * 4 + 3] * w;
    }
    out[2 * gid]     = U / l;                        // pseudo u (col)
    out[2 * gid + 1] = V / l;                        // pseudo v (row)
    // match_weights = keypoint_scores[tgt_ids] (S=1)
    out[2 * NPAIR * NKP + gid] = kp_scores[(size_t)(2 * p + 1) * NKP + qv];
    if (gid == 0) {                                  // tgt_ids, src_ids as floats
        out[6144] = 1.0f; out[6145] = 3.0f;
        out[6146] = 0.0f; out[6147] = 2.0f;
    }
}

extern "C" void kernel_launch(void* const* d_in, const int* in_sizes, int n_in,
                              void* d_out, int out_size, void* d_ws, size_t ws_size,
                              hipStream_t stream) {
    const float* kp_scores = (const float*)d_in[0];   // (4,1,1024)
    const float* kp_desc   = (const float*)d_in[1];   // (4,256,1024)
    const float* dense     = (const float*)d_in[2];   // (4,256,320,320)
    // d_in[3] keypoint_coords: unused by reference outputs
    float* ws   = (float*)d_ws;
    float* qn   = ws + WS_QN;
    float* invk = ws + WS_INVK;
    float* part = ws + WS_PART;
    float* out  = (float*)d_out;

    hipLaunchKernelGGL(qnorm_kernel,   dim3(8),   dim3(256), 0, stream, kp_desc, qn);
    hipLaunchKernelGGL(knorm_kernel,   dim3(800), dim3(256), 0, stream, dense, invk);
    hipLaunchKernelGGL(matcher_main,   dim3(NPAIR * QG_PER_PAIR * NCHUNK), dim3(256), 0, stream,
                       dense, qn, invk, part);
    hipLaunchKernelGGL(matcher_reduce, dim3(8),   dim3(256), 0, stream, part, kp_scores, out);
}